// AdaptedEmbedding_7971459301961
// MI455X (gfx1250) — compile-verified
//
#include <hip/hip_runtime.h>

typedef __attribute__((ext_vector_type(2))) float v2f;
typedef __attribute__((ext_vector_type(8))) float v8f;

// LoRA-adapted embedding:
//   out[t, d] = emb[id_t, d] + 2.0 * sum_r lora_A[r, id_t] * lora_B[d, r]
// Fused gather + rank-16 GEMM using V_WMMA_F32_16X16X4_F32 (K accumulated 4x4).
__global__ __launch_bounds__(256) void lora_embed_wmma_kernel(
    const int*   __restrict__ ids,    // [16384]
    const float* __restrict__ emb,    // [VOCAB, DIM]
    const float* __restrict__ loraA,  // [RANK, VOCAB]
    const float* __restrict__ loraB,  // [DIM, RANK]
    float*       __restrict__ out)    // [16384, DIM]
{
    constexpr int    DIM     = 1024;
    constexpr int    VOCAB   = 50257;
    constexpr int    RANK    = 16;
    constexpr float  SCALING = 2.0f;   // ALPHA / R = 32 / 16

    const int lane = threadIdx.x & 31;
    const int wave = threadIdx.x >> 5;   // 0..7
    const int mlo  = lane & 15;          // M (for A) / N (for B,C,D) low index
    const int hi   = lane >> 4;          // lane half selects K pair / M+8
    const int m_base = blockIdx.x * 16;  // 16 tokens per block

    // Token id for M = lane&15 (all lanes load; lanes 0-15 and 16-31 mirror).
    const int my_id = ids[m_base + mlo];

    // ---- A operand: 16x16 (M x K) fp32 tile as 4 chunks of 16x4 ----
    // 32-bit A layout: lane L holds M=L&15; VGPR0=K even, VGPR1=K odd of the
    // pair selected by the lane half:  K = 4*c + 2*hi + {0,1}.
    // Fold SCALING into A so D = (2*A) x B + C_emb.
    v2f a[4];
    const int kb = hi * 2;
#pragma unroll
    for (int c = 0; c < 4; ++c) {
        const int k0 = 4 * c + kb;
        a[c].x = SCALING * loraA[(size_t)(k0 + 0) * VOCAB + my_id];
        a[c].y = SCALING * loraA[(size_t)(k0 + 1) * VOCAB + my_id];
    }

    // ids of the rows this lane accumulates in the C/D layout: M = v + 8*hi.
    // Lanes 0-15 hold id for M=lane, so broadcast with ds_bpermute (wave32).
    int row_id[8];
#pragma unroll
    for (int v = 0; v < 8; ++v)
        row_id[v] = __builtin_amdgcn_ds_bpermute((v + 8 * hi) << 2, my_id);

    // ---- each wave covers 8 N-tiles (128 columns), reusing A registers ----
#pragma unroll
    for (int t = 0; t < 8; ++t) {
        const int n_tile = wave * 8 + t;
        const int col = n_tile * 16 + mlo;         // absolute output column N

        // C operand = embedding gather, directly in WMMA C/D layout:
        // VGPR v <-> row M = v + 8*hi; lanes stripe 16 consecutive columns,
        // so each load touches two contiguous 64B row segments.
        v8f acc;
#pragma unroll
        for (int v = 0; v < 8; ++v)
            acc[v] = emb[(size_t)row_id[v] * DIM + col];

        // B operand: 16x16 (K x N) tile of lora_B^T. Lane holds N=col;
        // per chunk it needs the contiguous pair lora_B[col, 4c+kb .. +1].
        const float* brow = loraB + (size_t)col * RANK;
#pragma unroll
        for (int c = 0; c < 4; ++c) {
            v2f b;
            b.x = brow[4 * c + kb + 0];
            b.y = brow[4 * c + kb + 1];
            // D = A x B + C   (fp32 WMMA, exact reference math)
            acc = __builtin_amdgcn_wmma_f32_16x16x4_f32(
                /*neg_a=*/false, a[c], /*neg_b=*/false, b,
                /*c_mod=*/(short)0, acc, /*reuse_a=*/false, /*reuse_b=*/false);
        }

        // Store D (same layout as C): two contiguous 64B segments per store.
#pragma unroll
        for (int v = 0; v < 8; ++v)
            out[(size_t)(m_base + v + 8 * hi) * DIM + col] = acc[v];
    }
}

extern "C" void kernel_launch(void* const* d_in, const int* in_sizes, int n_in,
                              void* d_out, int out_size, void* d_ws, size_t ws_size,
                              hipStream_t stream) {
    const int*   ids   = (const int*)  d_in[0];   // input_ids [4,4096]
    const float* emb   = (const float*)d_in[1];   // emb_weight [50257,1024]
    const float* loraA = (const float*)d_in[2];   // lora_A [16,50257]
    const float* loraB = (const float*)d_in[3];   // lora_B [1024,16]
    float* out = (float*)d_out;                   // [4,4096,1024] fp32

    const int n_tokens = in_sizes[0];             // 16384
    const int m_tiles  = n_tokens / 16;           // 1024 blocks, 8 waves each

    lora_embed_wmma_kernel<<<m_tiles, 256, 0, stream>>>(ids, emb, loraA, loraB, out);
}